// LogSlaterDeterminant_9285719294231
// MI455X (gfx1250) — compile-verified
//
#include <hip/hip_runtime.h>
#include <hip/hip_bf16.h>

typedef __attribute__((ext_vector_type(2))) float v2f;
typedef __attribute__((ext_vector_type(8))) float v8f;

#define N_ORB 512
#define NMAT 64      // matrix dimension (NUP == NDOWN == 64)
#define NB   16      // LU panel width
#define LDA  68      // LDS row stride in floats (keeps float4 alignment, staggers banks)

// Wave-synchronous LDS programming: LDS ops from one wave are processed in
// order by the DS unit, so a compiler-level barrier is all that's required
// between cross-lane LDS phases within a wave.
__device__ __forceinline__ void wave_sync() {
    asm volatile("" ::: "memory");
    __builtin_amdgcn_wave_barrier();
}

__global__ __launch_bounds__(64)
void logslater_lu_kernel(const int* __restrict__ n,
                         const float* __restrict__ Mup,
                         const float* __restrict__ Mdown,
                         float* __restrict__ out) {
    __shared__ float sA[2][NMAT * LDA];      // 2 * 17408 B = 34816 B
    __shared__ int   sRidx[2][NMAT];
    __shared__ float sRes[2][2];             // {logabs, signflip} per wave

    const int b    = blockIdx.x;
    const int wave = threadIdx.x >> 5;       // 0 = up, 1 = down
    const int lane = threadIdx.x & 31;

    const float* __restrict__ M = (wave == 0) ? Mup : Mdown;
    const int* __restrict__ np  = n + (size_t)b * (2 * N_ORB) + wave * N_ORB;
    float* A = sA[wave];

    // ---- 1. Extract occupied orbital indices (stable ascending) ----------
    // Each lane owns 16 contiguous entries of the 512-wide occupation vector.
    int cnt = 0;
    #pragma unroll
    for (int t = 0; t < 16; ++t)
        cnt += (np[lane * 16 + t] != 0) ? 1 : 0;

    // inclusive wave32 prefix scan -> exclusive offset
    int pre = cnt;
    #pragma unroll
    for (int d = 1; d < 32; d <<= 1) {
        int o = __shfl_up(pre, d, 32);
        if (lane >= d) pre += o;
    }
    int w = pre - cnt;
    for (int t = 0; t < 16; ++t) {
        int idx = lane * 16 + t;
        if (np[idx] != 0) sRidx[wave][w++] = idx;
    }
    wave_sync();

    // ---- 2. Gather 64 selected rows of M into LDS (float4 loads) ---------
    const float4* __restrict__ Mv = reinterpret_cast<const float4*>(M);
    #pragma unroll
    for (int it = 0; it < 32; ++it) {
        int t = lane + 32 * it;              // 1024 float4 tasks
        int r = t >> 4;                      // row in A
        int q = t & 15;                      // float4 column chunk
        int src = sRidx[wave][r];
        float4 v = Mv[src * (NMAT / 4) + q];
        *reinterpret_cast<float4*>(&A[r * LDA + q * 4]) = v;
    }
    wave_sync();

    // ---- 3. Blocked LU with partial pivoting, NB = 16 ---------------------
    float logabs = 0.0f;                     // replicated across lanes
    int   neg    = 0;

    for (int kb = 0; kb < NMAT / NB; ++kb) {
        const int c0 = kb * NB;

        // ---- 3a. Panel factorization (columns c0 .. c0+15) ----
        for (int j = c0; j < c0 + NB; ++j) {
            // pivot search on column j, rows j..63 (2 rows per lane)
            float best = -1.0f;
            int   bi   = j;
            for (int i = j + lane; i < NMAT; i += 32) {
                float a = fabsf(A[i * LDA + j]);
                if (a > best) { best = a; bi = i; }
            }
            #pragma unroll
            for (int d = 16; d >= 1; d >>= 1) {
                float ob = __shfl_xor(best, d, 32);
                int   oi = __shfl_xor(bi, d, 32);
                if (ob > best || (ob == best && oi < bi)) { best = ob; bi = oi; }
            }
            const int p = bi;                // uniform across lanes
            if (p != j) {
                neg ^= 1;
                for (int c = lane; c < NMAT; c += 32) {
                    float t0 = A[j * LDA + c];
                    float t1 = A[p * LDA + c];
                    A[j * LDA + c] = t1;
                    A[p * LDA + c] = t0;
                }
            }
            wave_sync();

            const float d0 = A[j * LDA + j];
            logabs += __logf(fabsf(d0));
            neg ^= (d0 < 0.0f) ? 1 : 0;
            const float rinv = 1.0f / d0;

            // scale L column + rank-1 update restricted to panel columns
            for (int i = j + 1 + lane; i < NMAT; i += 32) {
                float l = A[i * LDA + j] * rinv;
                A[i * LDA + j] = l;
                for (int c = j + 1; c < c0 + NB; ++c)
                    A[i * LDA + c] -= l * A[j * LDA + c];
            }
            wave_sync();
        }

        if (kb == NMAT / NB - 1) break;

        // ---- 3b. Triangular solve: U12 = L11^{-1} * A12 ----
        // Forward substitution; column partition is lane-stable so only the
        // uniform panel reads cross lanes (panel data is already final).
        for (int s = c0; s < c0 + NB; ++s) {
            for (int r = s + 1; r < c0 + NB; ++r) {
                float lrs = A[r * LDA + s];
                for (int c = c0 + NB + lane; c < NMAT; c += 32)
                    A[r * LDA + c] -= lrs * A[s * LDA + c];
            }
        }
        wave_sync();

        // ---- 3c. Schur update: A22 -= L21 * U12 via V_WMMA_F32_16X16X4_F32
        // 16x16 tiles, K=16 as 4 chained k-steps of K=4. A-fragment negated
        // on load so D = (-L)*U + C performs the subtraction in f32.
        const int half = (lane >> 4);        // 0: lanes 0-15, 1: lanes 16-31
        const int lmod = (lane & 15);
        for (int i0 = c0 + NB; i0 < NMAT; i0 += 16) {
            for (int j0 = c0 + NB; j0 < NMAT; j0 += 16) {
                // load C tile: M = v + 8*half, N = lmod
                v8f cfr;
                const int mBase = i0 + 8 * half;
                const int nn    = j0 + lmod;
                #pragma unroll
                for (int v = 0; v < 8; ++v)
                    cfr[v] = A[(mBase + v) * LDA + nn];

                #pragma unroll
                for (int ks = 0; ks < 4; ++ks) {
                    const int kc = c0 + 4 * ks;
                    // A-fragment (16x4): M = lmod, K = v + 2*half
                    v2f afr;
                    const int am = i0 + lmod;
                    const int ak = kc + 2 * half;
                    afr.x = -A[am * LDA + ak];
                    afr.y = -A[am * LDA + ak + 1];
                    // B-fragment (4x16): N = lmod, K = v + 2*half
                    v2f bfr;
                    const int bk = kc + 2 * half;
                    bfr.x = A[bk * LDA + j0 + lmod];
                    bfr.y = A[(bk + 1) * LDA + j0 + lmod];

                    cfr = __builtin_amdgcn_wmma_f32_16x16x4_f32(
                        /*neg_a=*/false, afr, /*neg_b=*/false, bfr,
                        /*c_mod=*/(short)0, cfr,
                        /*reuse_a=*/false, /*reuse_b=*/false);
                }

                #pragma unroll
                for (int v = 0; v < 8; ++v)
                    A[(mBase + v) * LDA + nn] = cfr[v];
            }
        }
        wave_sync();
    }

    // ---- 4. Combine up/down and emit complex64 result ---------------------
    if (lane == 0) {
        sRes[wave][0] = logabs;
        sRes[wave][1] = (float)neg;
    }
    __syncthreads();
    if (threadIdx.x == 0) {
        float re = sRes[0][0] + sRes[1][0];
        int flip = ((int)sRes[0][1]) ^ ((int)sRes[1][1]);
        out[2 * b + 0] = re;
        out[2 * b + 1] = flip ? 3.14159265358979323846f : 0.0f;
    }
}

extern "C" void kernel_launch(void* const* d_in, const int* in_sizes, int n_in,
                              void* d_out, int out_size, void* d_ws, size_t ws_size,
                              hipStream_t stream) {
    const int*   n     = (const int*)d_in[0];
    const float* Mup   = (const float*)d_in[1];
    const float* Mdown = (const float*)d_in[2];
    float*       out   = (float*)d_out;

    const int batch = in_sizes[0] / (2 * N_ORB);   // n is (BATCH, 1024) int32

    dim3 grid(batch);
    dim3 block(64);                                 // 2 waves: up + down
    logslater_lu_kernel<<<grid, block, 0, stream>>>(n, Mup, Mdown, out);
}